// VOXCPM_FEAT_DECODER_403726926222
// MI455X (gfx1250) — compile-verified
//
#include <hip/hip_runtime.h>

typedef __attribute__((ext_vector_type(2))) float v2f;
typedef __attribute__((ext_vector_type(8))) float v8f;
typedef __attribute__((__vector_size__(4 * sizeof(int)))) int i32x4_;

#define H_      1024
#define NKV_    2
#define NG_     8
#define HD_     64
#define QLEN_   9
#define NL_     8
#define FF_     4096
#define FEAT_   64
#define MROWS_  18          // 2 * QLEN

#define KC_     64          // K-chunk staged per iteration
#define AP_     68          // A tile row pitch (floats): conflict-free b64 reads
#define WP_     136         // W tile row pitch (floats): half-waves on disjoint banks

#if defined(__has_builtin)
#if __has_builtin(__builtin_amdgcn_global_load_async_to_lds_b128) && \
    __has_builtin(__builtin_amdgcn_s_wait_asynccnt)
#define ASYNC_LDS_ 1
#endif
#endif

#ifdef ASYNC_LDS_
typedef __attribute__((address_space(1))) i32x4_* gas1_v4i_;
typedef __attribute__((address_space(3))) i32x4_* las3_v4i_;
#endif

// ---------------------------------------------------------------------------
// Skinny GEMM:  D[Mv x N] = (R ? R : 0) + A[Mv x K] @ W[K x N]
// M padded to 32. 8 waves/block, one 32x16 output tile per wave (two fp32
// 16x16 WMMA accumulators), 128 N-columns per block. Weights staged into LDS
// with coalesced 128-bit (async when available) copies; inner loop is pure
// ds_load + v_wmma_f32_16x16x4_f32.
// ---------------------------------------------------------------------------
template <int K, int N>
__global__ __launch_bounds__(256)
void gemm_wmma(const float* __restrict__ A, const float* __restrict__ W,
               const float* __restrict__ R, float* __restrict__ D, int Mv)
{
    __shared__ float As[32][AP_];     //  8.7 KB
    __shared__ float Ws[KC_][WP_];    // 34.8 KB

    const int tid   = threadIdx.x;
    const int wave  = tid >> 5;
    const int lane  = tid & 31;
    const int nbase = blockIdx.x * 128;          // block's N window
    const int col   = lane & 15;                 // N within 16-wide frag
    const int row   = lane & 15;                 // M within 16-tall frag
    const int k0    = (lane >> 4) * 2;           // K sub-offset (ISA layout)
    const int half  = lane >> 4;
    const int wcol  = wave * 16 + col;           // column inside LDS W tile

    v8f acc0 = {};   // rows 0..15
    v8f acc1 = {};   // rows 16..31

    for (int kc = 0; kc < K; kc += KC_) {
        // ---- stage A chunk (32 x KC_), zero-padding rows >= Mv -------------
        for (int idx = tid; idx < 32 * KC_; idx += 256) {
            const int r  = idx >> 6;             // KC_ == 64
            const int kk = idx & (KC_ - 1);
            As[r][kk] = (r < Mv) ? A[(long)r * K + kc + kk] : 0.0f;
        }
        // ---- stage W chunk (KC_ x 128) with 128-bit copies -----------------
        #pragma unroll
        for (int p = 0; p < (KC_ * 128) / (256 * 4); ++p) {   // 8 float4/thread
            const int idx4 = tid + p * 256;
            const int r    = idx4 >> 5;                       // 32 float4 / row
            const int c4   = (idx4 & 31) * 4;
            const float* src = W + (long)(kc + r) * N + nbase + c4;
#ifdef ASYNC_LDS_
            __builtin_amdgcn_global_load_async_to_lds_b128(
                (gas1_v4i_)(unsigned long long)src,
                (las3_v4i_)&Ws[r][c4],
                0, 0);
#else
            const float4 t = *(const float4*)src;
            *(float4*)&Ws[r][c4] = t;
#endif
        }
#ifdef ASYNC_LDS_
        __builtin_amdgcn_s_wait_asynccnt(0);
#endif
        __syncthreads();

        // ---- prefetch next weight chunk while we compute -------------------
        if (kc + KC_ < K) {
            const int pr = tid >> 2;                 // 64 rows
            const int ps = (tid & 3) * 32;           // 4 x 128B segments / row
            __builtin_prefetch(W + (long)(kc + KC_ + pr) * N + nbase + ps, 0, 0);
        }

        // ---- WMMA inner loop: pure LDS reads -------------------------------
        #pragma unroll
        for (int kb = 0; kb < KC_; kb += 4) {
            v2f a0, a1, b;
            a0.x = As[row][kb + k0];
            a0.y = As[row][kb + k0 + 1];
            a1.x = As[row + 16][kb + k0];
            a1.y = As[row + 16][kb + k0 + 1];
            b.x  = Ws[kb + k0][wcol];
            b.y  = Ws[kb + k0 + 1][wcol];
            acc0 = __builtin_amdgcn_wmma_f32_16x16x4_f32(
                       false, a0, false, b, (short)0, acc0, false, false);
            acc1 = __builtin_amdgcn_wmma_f32_16x16x4_f32(
                       false, a1, false, b, (short)0, acc1, false, false);
        }
        __syncthreads();
    }

    // ---- epilogue: optional residual add, guarded rows ---------------------
    #pragma unroll
    for (int i = 0; i < 8; ++i) {
        const int m0 = i + half * 8;
        const int m1 = m0 + 16;
        if (m0 < Mv) {
            const long idx = (long)m0 * N + nbase + wave * 16 + col;
            D[idx] = (R ? R[idx] : 0.0f) + acc0[i];
        }
        if (m1 < Mv) {
            const long idx = (long)m1 * N + nbase + wave * 16 + col;
            D[idx] = (R ? R[idx] : 0.0f) + acc1[i];
        }
    }
}

// ---------------------------------------------------------------------------
// Unit-norm each row:  out[r,:] = in[r,:] * rsqrt(sum(in[r,:]^2))
// ---------------------------------------------------------------------------
__global__ __launch_bounds__(256)
void unit_norm_rows(const float* __restrict__ in, float* __restrict__ out, int cols)
{
    __shared__ float red[256];
    const int r = blockIdx.x;
    const float* p = in + (long)r * cols;
    float s = 0.0f;
    for (int c = threadIdx.x; c < cols; c += 256) { float v = p[c]; s += v * v; }
    red[threadIdx.x] = s;
    __syncthreads();
    for (int off = 128; off > 0; off >>= 1) {
        if (threadIdx.x < off) red[threadIdx.x] += red[threadIdx.x + off];
        __syncthreads();
    }
    const float rs = rsqrtf(red[0]);
    float* o = out + (long)r * cols;
    for (int c = threadIdx.x; c < cols; c += 256) o[c] = p[c] * rs;
}

// ---------------------------------------------------------------------------
// Assemble h (18 x 1024): [dit+t | t ; feat_cond ; random @ in_proj_w (x2)]
// ---------------------------------------------------------------------------
__global__ __launch_bounds__(256)
void prep_h(const int* __restrict__ step, const float* __restrict__ random,
            const float* __restrict__ dit, const float* __restrict__ feat,
            const float* __restrict__ t_all, const float* __restrict__ in_proj_w,
            float* __restrict__ h)
{
    const int r = blockIdx.x;          // 0..17
    const int b = r / QLEN_, t = r % QLEN_;
    const float* tv = t_all + (long)(*step) * H_;
    float* out = h + (long)r * H_;
    if (t == 0) {
        for (int c = threadIdx.x; c < H_; c += 256)
            out[c] = (b == 0 ? dit[c] : 0.0f) + tv[c];
    } else if (t < 5) {
        const float* f = feat + (long)(b * 4 + (t - 1)) * H_;
        for (int c = threadIdx.x; c < H_; c += 256) out[c] = f[c];
    } else {
        const float* xr = random + (long)(t - 5) * FEAT_;
        for (int c = threadIdx.x; c < H_; c += 256) {
            float acc = 0.0f;
            for (int k = 0; k < FEAT_; ++k) acc += xr[k] * in_proj_w[(long)k * H_ + c];
            out[c] = acc;
        }
    }
}

// ---------------------------------------------------------------------------
// Attention: 32 heads (b, kv, g), one wave per head. RoPE, 9x9 softmax, AV.
// ---------------------------------------------------------------------------
__global__ __launch_bounds__(128)
void attn_kernel(const float* __restrict__ qkv, const float* __restrict__ cosq,
                 const float* __restrict__ sinq, float* __restrict__ aout)
{
    __shared__ float sh[4][3][QLEN_][HD_];   // q' / k' / v per head
    const int w    = threadIdx.x >> 5;
    const int lane = threadIdx.x & 31;
    const int head = blockIdx.x * 4 + w;
    const int b  = head >> 4;
    const int kv = (head >> 3) & 1;
    const int g  = head & 7;
    const int qoff = (kv * NG_ + g) * HD_;
    const int koff = H_ + kv * HD_;
    const int voff = H_ + NKV_ * HD_ + kv * HD_;
    const int QKVN = H_ + 2 * NKV_ * HD_;    // 1280

    for (int t = 0; t < QLEN_; ++t) {
        const float* rowp = qkv + (long)(b * QLEN_ + t) * QKVN;
        const float c0 = cosq[t * HD_ + lane],      c1 = cosq[t * HD_ + lane + 32];
        const float s0 = sinq[t * HD_ + lane],      s1 = sinq[t * HD_ + lane + 32];
        const float q0 = rowp[qoff + lane],         q1 = rowp[qoff + lane + 32];
        sh[w][0][t][lane]      = q0 * c0 + q1 * s0;   // sin already half-negated
        sh[w][0][t][lane + 32] = q1 * c1 + q0 * s1;
        const float k0 = rowp[koff + lane],         k1 = rowp[koff + lane + 32];
        sh[w][1][t][lane]      = k0 * c0 + k1 * s0;
        sh[w][1][t][lane + 32] = k1 * c1 + k0 * s1;
        sh[w][2][t][lane]      = rowp[voff + lane];
        sh[w][2][t][lane + 32] = rowp[voff + lane + 32];
    }
    __syncthreads();

    if (lane < QLEN_) {
        const int i = lane;
        float s[QLEN_];
        float mx = -3.4e38f;
        for (int j = 0; j < QLEN_; ++j) {
            float acc = 0.0f;
            for (int d = 0; d < HD_; ++d) acc += sh[w][0][i][d] * sh[w][1][j][d];
            s[j] = acc; mx = fmaxf(mx, acc);
        }
        float sum = 0.0f;
        for (int j = 0; j < QLEN_; ++j) { s[j] = __expf(s[j] - mx); sum += s[j]; }
        const float inv = 1.0f / sum;
        float* orow = aout + (long)(b * QLEN_ + i) * H_ + qoff;
        for (int d = 0; d < HD_; ++d) {
            float acc = 0.0f;
            for (int j = 0; j < QLEN_; ++j) acc += s[j] * sh[w][2][j][d];
            orow[d] = acc * inv;
        }
    }
}

// ---------------------------------------------------------------------------
// act = silu(gate) * up, from gu = [gate | up]  (rows x 2*FF)
// ---------------------------------------------------------------------------
__global__ __launch_bounds__(256)
void silu_mul(const float* __restrict__ gu, float* __restrict__ act, int rows)
{
    const long i = (long)blockIdx.x * 256 + threadIdx.x;
    const long total = (long)rows * FF_;
    if (i >= total) return;
    const long r = i / FF_, c = i % FF_;
    const float g = gu[r * 2 * FF_ + c];
    const float u = gu[r * 2 * FF_ + FF_ + c];
    act[i] = (g / (1.0f + __expf(-g))) * u;
}

// ---------------------------------------------------------------------------
// Final: unit-norm suffix rows, est = hs @ out_proj, CFG combine, Euler step.
// ---------------------------------------------------------------------------
__global__ __launch_bounds__(256)
void final_kernel(const int* __restrict__ step, const float* __restrict__ random,
                  const float* __restrict__ cfg, const float* __restrict__ cfgm,
                  const float* __restrict__ dt_all, const float* __restrict__ h,
                  const float* __restrict__ Wout, float* __restrict__ out,
                  int out_size)
{
    __shared__ float shs[8 * H_];     // 32 KB
    __shared__ float sest[512];
    __shared__ float red[256];
    const int tid = threadIdx.x;

    for (int r = 0; r < 8; ++r) {
        const int gr = (r / 4) * QLEN_ + 5 + (r & 3);   // b0 t5..8, b1 t5..8
        const float* p = h + (long)gr * H_;
        float s = 0.0f;
        for (int c = tid; c < H_; c += 256) { float v = p[c]; s += v * v; }
        red[tid] = s; __syncthreads();
        for (int off = 128; off > 0; off >>= 1) {
            if (tid < off) red[tid] += red[tid + off];
            __syncthreads();
        }
        const float rs = rsqrtf(red[0]);
        for (int c = tid; c < H_; c += 256) shs[r * H_ + c] = p[c] * rs;
        __syncthreads();
    }

    for (int o = tid; o < 512; o += 256) {
        const int r = o >> 6, c = o & 63;
        float acc = 0.0f;
        for (int k = 0; k < H_; ++k) acc += shs[r * H_ + k] * Wout[(long)k * FEAT_ + c];
        sest[o] = acc;
    }
    __syncthreads();

    const float p = sest[tid];          // pos (batch 0), 256 values
    const float n = sest[256 + tid];    // neg (batch 1)
    red[tid] = p * n; __syncthreads();
    for (int off = 128; off > 0; off >>= 1) { if (tid < off) red[tid] += red[tid + off]; __syncthreads(); }
    const float pn = red[0]; __syncthreads();
    red[tid] = n * n; __syncthreads();
    for (int off = 128; off > 0; off >>= 1) { if (tid < off) red[tid] += red[tid + off]; __syncthreads(); }
    const float nn = red[0];

    const float st   = pn / nn;
    const float dt   = dt_all[*step];
    const float dphi = (*cfgm) * n * st + (*cfg) * p;
    const int off0 = (out_size > 256) ? 1 : 0;
    if (off0 && tid == 0) out[0] = (float)(*step + 1);
    out[off0 + tid] = random[tid] - dt * dphi;
}

// ---------------------------------------------------------------------------
extern "C" void kernel_launch(void* const* d_in, const int* in_sizes, int n_in,
                              void* d_out, int out_size, void* d_ws, size_t ws_size,
                              hipStream_t stream)
{
    (void)in_sizes; (void)n_in; (void)ws_size;
    const int*   step    = (const int*)  d_in[0];
    const float* random  = (const float*)d_in[1];
    const float* dit     = (const float*)d_in[2];
    const float* feat    = (const float*)d_in[3];
    const float* cfg     = (const float*)d_in[4];
    const float* cfgm    = (const float*)d_in[5];
    const float* t_all   = (const float*)d_in[6];
    const float* dt_all  = (const float*)d_in[7];
    const float* cosq    = (const float*)d_in[8];
    const float* sinq    = (const float*)d_in[9];
    const float* in_proj = (const float*)d_in[10];
    const float* out_proj= (const float*)d_in[11];
    const float* qkv_w   = (const float*)d_in[12];
    const float* o_w     = (const float*)d_in[13];
    const float* gu_w    = (const float*)d_in[14];
    const float* dn_w    = (const float*)d_in[15];

    float* ws   = (float*)d_ws;
    float* h    = ws;                        // 18*1024
    float* hn   = h    + MROWS_ * H_;        // 18*1024
    float* qkv  = hn   + MROWS_ * H_;        // 18*1280
    float* aout = qkv  + MROWS_ * 1280;      // 18*1024
    float* gu   = aout + MROWS_ * H_;        // 18*8192
    float* act  = gu   + MROWS_ * 2 * FF_;   // 18*4096

    prep_h<<<MROWS_, 256, 0, stream>>>(step, random, dit, feat, t_all, in_proj, h);

    for (int l = 0; l < NL_; ++l) {
        unit_norm_rows<<<MROWS_, 256, 0, stream>>>(h, hn, H_);
        gemm_wmma<H_, 1280><<<1280 / 128, 256, 0, stream>>>(
            hn, qkv_w + (long)l * H_ * 1280, nullptr, qkv, MROWS_);
        attn_kernel<<<8, 128, 0, stream>>>(qkv, cosq, sinq, aout);
        gemm_wmma<H_, H_><<<H_ / 128, 256, 0, stream>>>(
            aout, o_w + (long)l * H_ * H_, h, h, MROWS_);
        unit_norm_rows<<<MROWS_, 256, 0, stream>>>(h, hn, H_);
        gemm_wmma<H_, 2 * FF_><<<(2 * FF_) / 128, 256, 0, stream>>>(
            hn, gu_w + (long)l * H_ * 2 * FF_, nullptr, gu, MROWS_);
        silu_mul<<<(MROWS_ * FF_ + 255) / 256, 256, 0, stream>>>(gu, act, MROWS_);
        gemm_wmma<FF_, H_><<<H_ / 128, 256, 0, stream>>>(
            act, dn_w + (long)l * FF_ * H_, h, h, MROWS_);
    }

    final_kernel<<<1, 256, 0, stream>>>(step, random, cfg, cfgm, dt_all, h,
                                        out_proj, (float*)d_out, out_size);
}